// TangentProject_43147241455995
// MI455X (gfx1250) — compile-verified
//
#include <hip/hip_runtime.h>
#include <math.h>
#include <stdint.h>

// ---------------------------------------------------------------------------
// TangentProject SPD: out[b] = symlog( R * X[b] * R ),  R = symrsqrt(weight)
// N = 128, B = 4096, fp32 throughout. WMMA path: v_wmma_f32_16x16x4_f32.
// Staging path: global_load_async_to_lds_b32 (ASYNCcnt).
// ---------------------------------------------------------------------------

#define NDIM 128
#define LDA  133          // odd pitch, coprime with 64 banks -> conflict-free column walks
#define PSI  0.001f

typedef __attribute__((ext_vector_type(2))) float v2f;
typedef __attribute__((ext_vector_type(8))) float v8f;

// Shared-memory partition (dynamic LDS):
//   sA   : NDIM*LDA floats   (X -> M -> diagonalized A)
//   sV   : NDIM*LDA floats   (T scratch during GEMMs, then eigenvectors V)
//   sc/ssn : 64 rotation cos/sin ; slam : 128 f(lambda) ; sp/sq : 64 ints each
#define SMEM_FLOATS (2*NDIM*LDA + 64 + 64 + 128)
#define SMEM_INTS   (64 + 64)
#define SMEM_BYTES  ((SMEM_FLOATS + SMEM_INTS) * 4)

// ---------------------------------------------------------------------------
// One 16-row band of a 128x128 fp32 GEMM on the wave's WMMA unit.
// A fragment (16x4, 2 VGPR): m = lane&15, K = k0 + 2*(lane>>4) + vgpr
// B fragment (4x16, 2 VGPR): n = lane&15, K same rule
// C/D (16x16, 8 VGPR): row = vgpr + 8*(lane>>4), col = lane&15
// ---------------------------------------------------------------------------
template <bool BTRANS, bool ASCALE>
__device__ __forceinline__ void gemm_rowband(const float* __restrict__ Asrc, int lda_,
                                             const float* __restrict__ Bsrc, int ldb_,
                                             const float* __restrict__ ascale,
                                             int wave, int lane, v8f* acc) {
    const int ln = lane & 15;
    const int kh = lane >> 4;
    const int r0 = wave * 16;
#pragma unroll
    for (int t = 0; t < 8; ++t) acc[t] = v8f{};

    for (int k0 = 0; k0 < NDIM; k0 += 4) {
        const int K = k0 + 2 * kh;
        v2f a;
        a.x = Asrc[(r0 + ln) * lda_ + K];
        a.y = Asrc[(r0 + ln) * lda_ + K + 1];
        if (ASCALE) { a.x *= ascale[K]; a.y *= ascale[K + 1]; }
#pragma unroll
        for (int tn = 0; tn < 8; ++tn) {
            v2f b;
            if (BTRANS) {   // B = Ssrc^T : element (K, n) = Ssrc[n][K]
                b.x = Bsrc[(tn * 16 + ln) * ldb_ + K];
                b.y = Bsrc[(tn * 16 + ln) * ldb_ + K + 1];
            } else {
                b.x = Bsrc[(K)     * ldb_ + tn * 16 + ln];
                b.y = Bsrc[(K + 1) * ldb_ + tn * 16 + ln];
            }
            acc[tn] = __builtin_amdgcn_wmma_f32_16x16x4_f32(
                false, a, false, b, (short)0, acc[tn], false, false);
        }
    }
}

// ---------------------------------------------------------------------------
// Parallel two-sided Jacobi eigensolver on LDS. Closed-form round-robin
// pairing (no LDS permutation state): round r of 127, pair k of 64:
//   k==0 : (127, r) ; else ((r+k) mod 127, (r-k+127) mod 127)
// 64 disjoint rotations per round, 3 barriers per round, 256 threads.
// On exit: A holds diag(lambda), V holds eigenvectors (columns).
// ---------------------------------------------------------------------------
__device__ void jacobi_sym(float* A, float* V, float* sc, float* ssn,
                           int* sp, int* sq, int tid, int sweeps) {
    for (int idx = tid; idx < NDIM * NDIM; idx += 256) {
        const int i = idx >> 7, j = idx & 127;
        V[i * LDA + j] = (i == j) ? 1.0f : 0.0f;
    }
    __syncthreads();

    for (int sw = 0; sw < sweeps; ++sw) {
        for (int r = 0; r < NDIM - 1; ++r) {
            // 1) 64 rotation angles from the current matrix (closed-form pairing)
            if (tid < 64) {
                int p, q;
                if (tid == 0) { p = NDIM - 1; q = r; }
                else {
                    p = (r + tid) % (NDIM - 1);
                    q = (r - tid + (NDIM - 1)) % (NDIM - 1);
                }
                if (p > q) { int t = p; p = q; q = t; }
                sp[tid] = p; sq[tid] = q;
                const float apq = A[p * LDA + q];
                float c = 1.0f, s = 0.0f;
                if (fabsf(apq) > 1e-12f) {
                    const float tau = (A[q * LDA + q] - A[p * LDA + p]) / (2.0f * apq);
                    const float t = ((tau >= 0.0f) ? 1.0f : -1.0f) /
                                    (fabsf(tau) + sqrtf(1.0f + tau * tau));
                    c = 1.0f / sqrtf(1.0f + t * t);
                    s = t * c;
                }
                sc[tid] = c; ssn[tid] = s;
            }
            __syncthreads();
            // 2) column rotations of A (pairs own disjoint columns -> race-free)
            for (int item = tid; item < 64 * NDIM; item += 256) {
                const int k = item >> 7, i = item & 127;
                const int p = sp[k], q = sq[k];
                const float c = sc[k], s = ssn[k];
                const float ap = A[i * LDA + p], aq = A[i * LDA + q];
                A[i * LDA + p] = c * ap - s * aq;
                A[i * LDA + q] = s * ap + c * aq;
            }
            __syncthreads();
            // 3) row rotations of A + accumulate V = V*J (disjoint rows/cols)
            for (int item = tid; item < 64 * NDIM; item += 256) {
                const int k = item >> 7, i = item & 127;
                const int p = sp[k], q = sq[k];
                const float c = sc[k], s = ssn[k];
                const float ap = A[p * LDA + i], aq = A[q * LDA + i];
                A[p * LDA + i] = c * ap - s * aq;
                A[q * LDA + i] = s * ap + c * aq;
                const float vp = V[i * LDA + p], vq = V[i * LDA + q];
                V[i * LDA + p] = c * vp - s * vq;
                V[i * LDA + q] = s * vp + c * vq;
            }
            __syncthreads();
        }
    }
}

// ---------------------------------------------------------------------------
// Kernel 1 (1 block): R = symmap(weight, rsqrt, PSI)  -> d_ws (128x128 f32)
// ---------------------------------------------------------------------------
__global__ __launch_bounds__(256, 1)
void TangentProject_prep_kernel(const float* __restrict__ W, float* __restrict__ Rout) {
    extern __shared__ float smem[];
    float* sA   = smem;
    float* sV   = sA + NDIM * LDA;
    float* sc   = sV + NDIM * LDA;
    float* ssn  = sc + 64;
    float* slam = ssn + 64;
    int*   sp   = (int*)(slam + NDIM);
    int*   sq   = sp + 64;

    const int tid  = threadIdx.x;
    const int lane = tid & 31;
    const int wave = tid >> 5;

    // load reconditioned weight: (1-psi)*W + psi*I  (recondition fused -> VGPR path)
    for (int idx = tid; idx < NDIM * NDIM; idx += 256) {
        const int i = idx >> 7, j = idx & 127;
        sA[i * LDA + j] = (1.0f - PSI) * W[idx] + ((i == j) ? PSI : 0.0f);
    }
    __syncthreads();

    jacobi_sym(sA, sV, sc, ssn, sp, sq, tid, 12);

    if (tid < NDIM) slam[tid] = rsqrtf(fmaxf(sA[tid * LDA + tid], 1e-30f));
    __syncthreads();

    // R = (V * diag(rsqrt(lam))) @ V^T   -> global, full fp32 WMMA
    v8f acc[8];
    gemm_rowband<true, true>(sV, LDA, sV, LDA, slam, wave, lane, acc);
    const int ln = lane & 15, kh = lane >> 4, r0 = wave * 16;
#pragma unroll
    for (int tn = 0; tn < 8; ++tn)
#pragma unroll
        for (int v = 0; v < 8; ++v)
            Rout[(r0 + v + 8 * kh) * NDIM + tn * 16 + ln] = acc[tn][v];
}

// ---------------------------------------------------------------------------
// Kernel 2 (one block per batch matrix):
//   T = R @ X ; M = (1-psi)*(T @ R) + psi*I ; eigh(M) ; out = V diag(log l) V^T
// ---------------------------------------------------------------------------
__global__ __launch_bounds__(256, 1)
void TangentProject_batch_kernel(const float* __restrict__ X,
                                 const float* __restrict__ Rg,
                                 float* __restrict__ out) {
    extern __shared__ float smem[];
    float* sA   = smem;              // X -> M -> diagonalized
    float* sV   = sA + NDIM * LDA;   // T scratch, then eigenvectors
    float* sc   = sV + NDIM * LDA;
    float* ssn  = sc + 64;
    float* slam = ssn + 64;
    int*   sp   = (int*)(slam + NDIM);
    int*   sq   = sp + 64;

    const int tid  = threadIdx.x;
    const int lane = tid & 31;
    const int wave = tid >> 5;
    const int ln   = lane & 15;
    const int kh   = lane >> 4;
    const int r0   = wave * 16;

    const float* Xg = X + (size_t)blockIdx.x * NDIM * NDIM;

    // warm L2/L0 for the shared whitener R while staging runs
    __builtin_prefetch((const char*)Rg + (tid << 6), 0, 0);

    // stage X into LDS via async DMA path (no VGPR round-trip, ASYNCcnt-tracked)
    for (int idx = tid; idx < NDIM * NDIM; idx += 256) {
        const int i = idx >> 7, j = idx & 127;
        const unsigned lds_addr = (unsigned)(uintptr_t)(sA + i * LDA + j);
        const unsigned goff     = (unsigned)(idx << 2);
        asm volatile("global_load_async_to_lds_b32 %0, %1, %2"
                     :: "v"(lds_addr), "v"(goff), "s"(Xg)
                     : "memory");
    }
    asm volatile("s_wait_asynccnt 0" ::: "memory");
    __syncthreads();

    v8f acc[8];

    // GEMM1:  T(sV) = R(global, L2-resident) @ X(sA)
    gemm_rowband<false, false>(Rg, NDIM, sA, LDA, nullptr, wave, lane, acc);
#pragma unroll
    for (int tn = 0; tn < 8; ++tn)
#pragma unroll
        for (int v = 0; v < 8; ++v)
            sV[(r0 + v + 8 * kh) * LDA + tn * 16 + ln] = acc[tn][v];
    __syncthreads();

    // GEMM2:  M(sA) = T(sV) @ R(global), psi-recondition fused into the store
    gemm_rowband<false, false>(sV, LDA, Rg, NDIM, nullptr, wave, lane, acc);
#pragma unroll
    for (int tn = 0; tn < 8; ++tn)
#pragma unroll
        for (int v = 0; v < 8; ++v) {
            const int row = r0 + v + 8 * kh;
            const int col = tn * 16 + ln;
            sA[row * LDA + col] = (1.0f - PSI) * acc[tn][v] + ((row == col) ? PSI : 0.0f);
        }
    __syncthreads();

    // batched symmetric eigensolve (VALU/LDS bound; no tensor path for rotations)
    jacobi_sym(sA, sV, sc, ssn, sp, sq, tid, 8);

    if (tid < NDIM) slam[tid] = logf(fmaxf(sA[tid * LDA + tid], 1e-30f));
    __syncthreads();

    // GEMM3:  out = (V * diag(log lam)) @ V^T, C-fragments streamed to HBM
    gemm_rowband<true, true>(sV, LDA, sV, LDA, slam, wave, lane, acc);
    float* og = out + (size_t)blockIdx.x * NDIM * NDIM;
#pragma unroll
    for (int tn = 0; tn < 8; ++tn)
#pragma unroll
        for (int v = 0; v < 8; ++v)
            og[(r0 + v + 8 * kh) * NDIM + tn * 16 + ln] = acc[tn][v];
}

// ---------------------------------------------------------------------------
extern "C" void kernel_launch(void* const* d_in, const int* in_sizes, int n_in,
                              void* d_out, int out_size, void* d_ws, size_t ws_size,
                              hipStream_t stream) {
    const float* x = (const float*)d_in[0];   // [B,128,128] fp32
    const float* w = (const float*)d_in[1];   // [128,128]   fp32
    float* out = (float*)d_out;
    float* R   = (float*)d_ws;                // 128*128 fp32 scratch

    const int batch = in_sizes[0] / (NDIM * NDIM);

    TangentProject_prep_kernel<<<1, 256, SMEM_BYTES, stream>>>(w, R);
    TangentProject_batch_kernel<<<batch, 256, SMEM_BYTES, stream>>>(x, R, out);
}